// MambaPhysicsModel_58007828299841
// MI455X (gfx1250) — compile-verified
//
#include <hip/hip_runtime.h>
#include <math.h>

#define B_SZ 32
#define LSEQ 251
#define DMODEL 512
#define DIN 1024
#define DSTATE 16
#define DTRANK 32
#define NLAYERS 4
#define BL (B_SZ * LSEQ) /* 8032 */
#define DTS 0.02f

#define BKC 32   /* K-chunk for TDM GEMM */
#define LDSTR 34 /* padded LDS row stride (TDM pad: +2 dwords per 32) */

typedef float v2f __attribute__((ext_vector_type(2)));
typedef float v8f __attribute__((ext_vector_type(8)));
typedef unsigned int v4u __attribute__((ext_vector_type(4)));
typedef unsigned int v8u __attribute__((ext_vector_type(8)));

__device__ __forceinline__ float siluf(float x) { return x / (1.f + expf(-x)); }

__device__ __forceinline__ float actf(float v, int act) {
  if (act == 1) return fmaxf(v, 0.f);
  if (act == 2) return (v > 20.f) ? v : log1pf(expf(v));
  return v;
}

// ---- TDM: async 2D tile load (rows x 32 f32) global->LDS, row-padded +2 dw ----
// Descriptor per CDNA5 ISA ch.8: group0 {count,lds,global,type=2}, group1 dims.
__device__ __forceinline__ void tdm_load_tile(unsigned int lds_off,
                                              const float* gptr,
                                              unsigned int rows,
                                              unsigned int row_stride_elems) {
  unsigned long long ga = (unsigned long long)(uintptr_t)gptr;
  v4u g0;
  g0[0] = 1u;                                   // count=1 valid descriptor
  g0[1] = lds_off;                              // lds_addr (bytes)
  g0[2] = (unsigned int)ga;                     // global_addr[31:0]
  g0[3] = (unsigned int)(ga >> 32) | (2u << 30);// global_addr[56:32] | type=2
  v8u g1;
  g1[0] = (2u << 16)      // data_size = 4 bytes
        | (1u << 20)      // pad_enable
        | (4u << 22)      // pad_interval: every 32 dwords
        | (1u << 25);     // pad_amount: 2 dwords
  g1[1] = (BKC & 0xFFFFu) << 16;                 // tensor_dim0 lo16 (=32)
  g1[2] = (rows & 0xFFFFu) << 16;                // dim0 hi16=0 | tensor_dim1 lo16
  g1[3] = (rows >> 16) | ((BKC & 0xFFFFu) << 16);// tensor_dim1 hi16 | tile_dim0=32
  g1[4] = rows & 0xFFFFu;                        // tile_dim1=rows | tile_dim2=0
  g1[5] = row_stride_elems;                      // tensor_dim0_stride lo32
  g1[6] = 0u;
  g1[7] = 0u;
  asm volatile("tensor_load_to_lds %0, %1" ::"s"(g0), "s"(g1) : "memory");
}

// ---------------- preprocess: gaussian smooth (sigma=2, reflect) + features ----
__global__ void k_preprocess(const float* __restrict__ temp, float* __restrict__ feats) {
  int b = blockIdx.x;
  int t = threadIdx.x;
  __shared__ float sTs[LSEQ];
  float w[17];
  float s = 0.f;
#pragma unroll
  for (int j = 0; j < 17; ++j) {
    float i = (float)(j - 8);
    w[j] = expf(-0.125f * i * i);
    s += w[j];
  }
  float inv = 1.f / s;
  if (t < LSEQ) {
    float acc = 0.f;
#pragma unroll
    for (int j = 0; j < 17; ++j) {
      int i = t + j - 8;
      if (i < 0) i = -1 - i;
      if (i >= LSEQ) i = 2 * LSEQ - 1 - i;
      acc += w[j] * temp[b * LSEQ + i];
    }
    sTs[t] = acc * inv;
  }
  __syncthreads();
  if (t < LSEQ) {
    float Ttil = sTs[t] - sTs[0];
    float Tdot = (t < LSEQ - 1) ? (sTs[t + 1] - sTs[t]) / DTS
                                : (sTs[LSEQ - 1] - sTs[LSEQ - 2]) / DTS;
    float tv = (t == 0) ? DTS : (float)t * DTS;
    float tinv = 1.f / sqrtf(tv);
    size_t base = ((size_t)b * LSEQ + t) * 4;
    feats[base + 0] = Ttil;
    feats[base + 1] = Tdot;
    feats[base + 2] = tinv;
    feats[base + 3] = DTS / 5.02f;
  }
}

// ---------------- stats = feats.mean over L ----------------
__global__ void k_stats(const float* __restrict__ feats, float* __restrict__ stats) {
  int i = threadIdx.x;
  if (i >= B_SZ * 4) return;
  int b = i >> 2, c = i & 3;
  float s = 0.f;
  for (int t = 0; t < LSEQ; ++t) s += feats[((size_t)b * LSEQ + t) * 4 + c];
  stats[i] = s / (float)LSEQ;
}

// ---------------- physics-direct MLP (4->64 relu ->32) ----------------
__global__ void k_pd(const float* __restrict__ stats, const float* __restrict__ w1,
                     const float* __restrict__ b1, const float* __restrict__ w2,
                     const float* __restrict__ b2, float* __restrict__ pd,
                     float* __restrict__ f) {
  int b = blockIdx.x;
  int j = threadIdx.x;
  __shared__ float sh[64];
  float s = b1[j];
  for (int c = 0; c < 4; ++c) s += w1[j * 4 + c] * stats[b * 4 + c];
  sh[j] = fmaxf(s, 0.f);
  __syncthreads();
  if (j < 32) {
    float v = b2[j];
    for (int k = 0; k < 64; ++k) v += w2[j * 64 + k] * sh[k];
    pd[b * 32 + j] = v;
    f[(size_t)b * (DMODEL + 32) + DMODEL + j] = v;
  }
}

// ---------------- input projection (K=4) + BatchNorm (eval) ----------------
__global__ void k_inp_bn(const float* __restrict__ feats, const float* __restrict__ iw,
                         const float* __restrict__ ib, const float* __restrict__ g,
                         const float* __restrict__ bb, const float* __restrict__ rm,
                         const float* __restrict__ rv, float* __restrict__ x) {
  int idx = blockIdx.x * blockDim.x + threadIdx.x;
  if (idx >= BL * DMODEL) return;
  int m = idx % DMODEL;
  int bt = idx / DMODEL;
  const float* f = feats + (size_t)bt * 4;
  float v = ib[m] + iw[m * 4 + 0] * f[0] + iw[m * 4 + 1] * f[1] +
            iw[m * 4 + 2] * f[2] + iw[m * 4 + 3] * f[3];
  v = (v - rm[m]) * rsqrtf(rv[m] + 1e-5f) * g[m] + bb[m];
  x[idx] = v;
}

// ============ TDM + LDS double-buffered WMMA GEMM ============
// C[M,N] = A[M,K](lda) @ W[N,K]^T (+bias)(act)
// grid: (N/128, M/32), 256 threads (8 waves), wave w -> 16x32 subtile.
// Requires M%32==0, N%128==0, K%32==0.
__global__ void __launch_bounds__(256) k_gemm_tdm(
    const float* __restrict__ A, int lda, const float* __restrict__ W,
    const float* __restrict__ bias, float* __restrict__ C, int N, int K, int act) {
  __shared__ float sA[2][32 * LDSTR];
  __shared__ float sB[2][128 * LDSTR];
  int tm = blockIdx.y * 32;
  int tn = blockIdx.x * 128;
  int wave = threadIdx.x >> 5;
  int lane = threadIdx.x & 31;
  int wrow = wave >> 2;  // 0..1 -> +16*wrow in M
  int wcol = wave & 3;   // 0..3 -> +32*wcol in N
  int half = lane >> 4;  // upper lanes carry K+2,K+3
  int l = lane & 15;

  unsigned int ldsA[2] = {(unsigned int)(uintptr_t)&sA[0][0],
                          (unsigned int)(uintptr_t)&sA[1][0]};
  unsigned int ldsB[2] = {(unsigned int)(uintptr_t)&sB[0][0],
                          (unsigned int)(uintptr_t)&sB[1][0]};
  const float* Ablk = A + (size_t)tm * lda;
  const float* Wblk = W + (size_t)tn * K;
  int nchunks = K / BKC;

  if (wave == 0) {  // wave-uniform branch: TDM issued once (EXEC ignored)
    tdm_load_tile(ldsA[0], Ablk, 32u, (unsigned int)lda);
    tdm_load_tile(ldsB[0], Wblk, 128u, (unsigned int)K);
    __builtin_amdgcn_s_wait_tensorcnt(0);
  }
  __syncthreads();

  v8f acc0 = {0.f, 0.f, 0.f, 0.f, 0.f, 0.f, 0.f, 0.f};
  v8f acc1 = {0.f, 0.f, 0.f, 0.f, 0.f, 0.f, 0.f, 0.f};
  for (int c = 0; c < nchunks; ++c) {
    int cur = c & 1;
    if (wave == 0 && c + 1 < nchunks) {
      int k0 = (c + 1) * BKC;
      tdm_load_tile(ldsA[cur ^ 1], Ablk + k0, 32u, (unsigned int)lda);
      tdm_load_tile(ldsB[cur ^ 1], Wblk + k0, 128u, (unsigned int)K);
    }
    const float* a_base = &sA[cur][(wrow * 16 + l) * LDSTR + 2 * half];
    const float* b0_base = &sB[cur][(wcol * 32 + l) * LDSTR + 2 * half];
    const float* b1_base = &sB[cur][(wcol * 32 + 16 + l) * LDSTR + 2 * half];
#pragma unroll
    for (int k = 0; k < BKC; k += 4) {
      v2f a, b0, b1;
      a.x = a_base[k];
      a.y = a_base[k + 1];
      b0.x = b0_base[k];
      b0.y = b0_base[k + 1];
      b1.x = b1_base[k];
      b1.y = b1_base[k + 1];
      acc0 = __builtin_amdgcn_wmma_f32_16x16x4_f32(false, a, false, b0, (short)0,
                                                   acc0, false, false);
      acc1 = __builtin_amdgcn_wmma_f32_16x16x4_f32(false, a, false, b1, (short)0,
                                                   acc1, false, false);
    }
    if (wave == 0 && c + 1 < nchunks) __builtin_amdgcn_s_wait_tensorcnt(0);
    __syncthreads();
  }

  int col0 = tn + wcol * 32 + l;
  int col1 = col0 + 16;
  float bv0 = bias ? bias[col0] : 0.f;
  float bv1 = bias ? bias[col1] : 0.f;
#pragma unroll
  for (int j = 0; j < 8; ++j) {
    int row = tm + wrow * 16 + half * 8 + j;
    C[(size_t)row * N + col0] = actf(acc0[j] + bv0, act);
    C[(size_t)row * N + col1] = actf(acc1[j] + bv1, act);
  }
}

// ---------------- direct WMMA fp32 GEMM (small-N cases) ----------------
__global__ void __launch_bounds__(256) k_gemm_wmma(
    const float* __restrict__ A, int lda, const float* __restrict__ W,
    const float* __restrict__ bias, float* __restrict__ C, int M, int N, int K,
    int act) {
  int wave = (blockIdx.x * blockDim.x + threadIdx.x) >> 5;
  int lane = threadIdx.x & 31;
  int mt = (M + 15) >> 4, nt = (N + 15) >> 4;
  if (wave >= mt * nt) return;  // wave-uniform: EXEC stays all-ones for WMMA
  int tm = (wave / nt) << 4;
  int tn = (wave % nt) << 4;
  int half = lane >> 4;
  int l = lane & 15;
  int rowA = tm + l;
  int colB = tn + l;
  bool cok = colB < N;
  const float* arow = A + (size_t)rowA * lda + 2 * half;
  const float* brow = W + (size_t)(cok ? colB : 0) * K + 2 * half;
  v8f acc = {0.f, 0.f, 0.f, 0.f, 0.f, 0.f, 0.f, 0.f};
  for (int k = 0; k < K; k += 4) {
    v2f a, b;
    a.x = arow[k];
    a.y = arow[k + 1];
    float b0 = brow[k];
    float b1 = brow[k + 1];
    b.x = cok ? b0 : 0.f;
    b.y = cok ? b1 : 0.f;
    acc = __builtin_amdgcn_wmma_f32_16x16x4_f32(false, a, false, b, (short)0,
                                                acc, false, false);
  }
  int col = tn + l;
  if (col < N) {
    float bv = bias ? bias[col] : 0.f;
#pragma unroll
    for (int j = 0; j < 8; ++j) {
      int row = tm + half * 8 + j;
      C[(size_t)row * N + col] = actf(acc[j] + bv, act);
    }
  }
}

// ---------------- causal depthwise conv1d (width 4) + SiLU ----------------
__global__ void k_conv_silu(const float* __restrict__ xz, const float* __restrict__ cw,
                            const float* __restrict__ cb, float* __restrict__ u) {
  int idx = blockIdx.x * blockDim.x + threadIdx.x;
  if (idx >= BL * DIN) return;
  int d = idx % DIN;
  int bt = idx / DIN;
  int t = bt % LSEQ;
  int b = bt / LSEQ;
  float s = cb[d];
#pragma unroll
  for (int j = 0; j < 4; ++j) {
    int tt = t - 3 + j;
    if (tt >= 0) s += cw[d * 4 + j] * xz[(size_t)(b * LSEQ + tt) * (2 * DIN) + d];
  }
  u[idx] = siluf(s);
}

// ---------------- selective scan: one thread per (b, channel) ----------------
__global__ void k_scan(const float* __restrict__ delta, const float* __restrict__ u,
                       const float* __restrict__ dbc, const float* __restrict__ Alog,
                       const float* __restrict__ Dp, float* __restrict__ y) {
  int b = blockIdx.y;
  int d = blockIdx.x * blockDim.x + threadIdx.x;
  __shared__ float sB[DSTATE], sC[DSTATE];
  float Ar[DSTATE];
#pragma unroll
  for (int s = 0; s < DSTATE; ++s) Ar[s] = -expf(Alog[d * DSTATE + s]);
  float Dd = Dp[d];
  float h[DSTATE];
#pragma unroll
  for (int s = 0; s < DSTATE; ++s) h[s] = 0.f;
  for (int t = 0; t < LSEQ; ++t) {
    size_t base = (size_t)b * LSEQ + t;
    if (threadIdx.x < 32) {
      float v = dbc[base * (DTRANK + 2 * DSTATE) + DTRANK + threadIdx.x];
      if (threadIdx.x < DSTATE) sB[threadIdx.x] = v;
      else sC[threadIdx.x - DSTATE] = v;
    }
    __syncthreads();
    float dl = delta[base * DIN + d];
    float ut = u[base * DIN + d];
    float du = dl * ut;
    float yv = 0.f;
#pragma unroll
    for (int s = 0; s < DSTATE; ++s) {
      h[s] = h[s] * expf(dl * Ar[s]) + du * sB[s];
      yv += h[s] * sC[s];
    }
    y[base * DIN + d] = yv + ut * Dd;
    __syncthreads();
  }
}

// ---------------- gating: y *= silu(z) ----------------
__global__ void k_gate(float* __restrict__ y, const float* __restrict__ xz) {
  int idx = blockIdx.x * blockDim.x + threadIdx.x;
  if (idx >= BL * DIN) return;
  int d = idx % DIN;
  int bt = idx / DIN;
  float z = xz[(size_t)bt * (2 * DIN) + DIN + d];
  y[idx] *= siluf(z);
}

// ---------------- residual add + LayerNorm (block per row of 512) ----------------
__global__ void k_resln(float* __restrict__ x, const float* __restrict__ o,
                        const float* __restrict__ g, const float* __restrict__ bb) {
  int row = blockIdx.x;
  int tid = threadIdx.x;
  __shared__ float red[256];
  size_t base = (size_t)row * DMODEL;
  float v0 = x[base + tid] + o[base + tid];
  float v1 = x[base + tid + 256] + o[base + tid + 256];
  red[tid] = v0 + v1;
  __syncthreads();
  for (int st = 128; st > 0; st >>= 1) {
    if (tid < st) red[tid] += red[tid + st];
    __syncthreads();
  }
  float mean = red[0] / (float)DMODEL;
  __syncthreads();
  float d0 = v0 - mean, d1 = v1 - mean;
  red[tid] = d0 * d0 + d1 * d1;
  __syncthreads();
  for (int st = 128; st > 0; st >>= 1) {
    if (tid < st) red[tid] += red[tid + st];
    __syncthreads();
  }
  float inv = rsqrtf(red[0] / (float)DMODEL + 1e-5f);
  x[base + tid] = d0 * inv * g[tid] + bb[tid];
  x[base + tid + 256] = d1 * inv * g[tid + 256] + bb[tid + 256];
}

// ---------------- mean-pool over L into fusion buffer ----------------
__global__ void k_pool(const float* __restrict__ x, float* __restrict__ f) {
  int idx = blockIdx.x * blockDim.x + threadIdx.x;
  if (idx >= B_SZ * DMODEL) return;
  int m = idx % DMODEL;
  int b = idx / DMODEL;
  float s = 0.f;
  for (int t = 0; t < LSEQ; ++t) s += x[((size_t)b * LSEQ + t) * DMODEL + m];
  f[(size_t)b * (DMODEL + 32) + m] = s / (float)LSEQ;
}

// ---------------- final: out = h2@out_w^T + out_b + 0.1*(pd@aux_w^T + aux_b) ----
__global__ void k_final(const float* __restrict__ h2, const float* __restrict__ pd,
                        const float* __restrict__ ow, const float* __restrict__ ob,
                        const float* __restrict__ aw, const float* __restrict__ ab,
                        float* __restrict__ out) {
  int b = threadIdx.x;
  if (b >= B_SZ) return;
  float s = ob[0];
  for (int k = 0; k < 64; ++k) s += h2[b * 64 + k] * ow[k];
  float a = ab[0];
  for (int k = 0; k < 32; ++k) a += pd[b * 32 + k] * aw[k];
  out[b] = s + 0.1f * a;
}

extern "C" void kernel_launch(void* const* d_in, const int* in_sizes, int n_in,
                              void* d_out, int out_size, void* d_ws, size_t ws_size,
                              hipStream_t stream) {
  (void)in_sizes; (void)n_in; (void)out_size; (void)ws_size;
  const float* temp = (const float*)d_in[0];
  const float* inp_w = (const float*)d_in[1];
  const float* inp_b = (const float*)d_in[2];
  const float* bn_g = (const float*)d_in[3];
  const float* bn_b = (const float*)d_in[4];
  const float* bn_rm = (const float*)d_in[5];
  const float* bn_rv = (const float*)d_in[6];
  const float* in_proj_w = (const float*)d_in[7];
  const float* conv_w = (const float*)d_in[8];
  const float* conv_b = (const float*)d_in[9];
  const float* x_proj_w = (const float*)d_in[10];
  const float* dt_proj_w = (const float*)d_in[11];
  const float* dt_proj_b = (const float*)d_in[12];
  const float* A_log = (const float*)d_in[13];
  const float* Dp = (const float*)d_in[14];
  const float* out_proj_w = (const float*)d_in[15];
  const float* ln_g = (const float*)d_in[16];
  const float* ln_b = (const float*)d_in[17];
  const float* pd_w1 = (const float*)d_in[18];
  const float* pd_b1 = (const float*)d_in[19];
  const float* pd_w2 = (const float*)d_in[20];
  const float* pd_b2 = (const float*)d_in[21];
  const float* f_w1 = (const float*)d_in[22];
  const float* f_b1 = (const float*)d_in[23];
  const float* f_w2 = (const float*)d_in[24];
  const float* f_b2 = (const float*)d_in[25];
  const float* out_w = (const float*)d_in[26];
  const float* out_b = (const float*)d_in[27];
  const float* aux_w = (const float*)d_in[28];
  const float* aux_b = (const float*)d_in[29];
  float* outp = (float*)d_out;

  float* ws = (float*)d_ws;
  size_t off = 0;
  auto alloc = [&](size_t n) { float* p = ws + off; off += n; return p; };
  float* feats = alloc((size_t)BL * 4);
  float* x = alloc((size_t)BL * DMODEL);
  float* xz = alloc((size_t)BL * 2 * DIN);
  float* u = alloc((size_t)BL * DIN);
  float* dbc = alloc((size_t)BL * (DTRANK + 2 * DSTATE));
  float* delta = alloc((size_t)BL * DIN);
  float* ybuf = alloc((size_t)BL * DIN);
  float* outb = alloc((size_t)BL * DMODEL);
  float* statsb = alloc(B_SZ * 4);
  float* pdb = alloc(B_SZ * 32);
  float* fbuf = alloc((size_t)B_SZ * (DMODEL + 32));
  float* h1 = alloc((size_t)B_SZ * 256);
  float* h2 = alloc((size_t)B_SZ * 64);

  k_preprocess<<<B_SZ, 256, 0, stream>>>(temp, feats);
  k_stats<<<1, 128, 0, stream>>>(feats, statsb);
  k_pd<<<B_SZ, 64, 0, stream>>>(statsb, pd_w1, pd_b1, pd_w2, pd_b2, pdb, fbuf);
  {
    int n = BL * DMODEL;
    k_inp_bn<<<(n + 255) / 256, 256, 0, stream>>>(feats, inp_w, inp_b, bn_g, bn_b,
                                                  bn_rm, bn_rv, x);
  }
  auto gemm_grid = [](int M, int N) {
    long tiles = (long)((M + 15) / 16) * ((N + 15) / 16);
    return (int)((tiles + 7) / 8);
  };
  for (int layer = 0; layer < NLAYERS; ++layer) {
    // in_proj: [8032,512] @ [2048,512]^T -> xz   (TDM path)
    k_gemm_tdm<<<dim3((2 * DIN) / 128, BL / 32), 256, 0, stream>>>(
        x, DMODEL, in_proj_w + (size_t)layer * 2 * DIN * DMODEL, nullptr, xz,
        2 * DIN, DMODEL, 0);
    {
      int n = BL * DIN;
      k_conv_silu<<<(n + 255) / 256, 256, 0, stream>>>(
          xz, conv_w + (size_t)layer * DIN * 4, conv_b + (size_t)layer * DIN, u);
    }
    // x_proj: N=64 -> direct WMMA path
    k_gemm_wmma<<<gemm_grid(BL, DTRANK + 2 * DSTATE), 256, 0, stream>>>(
        u, DIN, x_proj_w + (size_t)layer * (DTRANK + 2 * DSTATE) * DIN, nullptr,
        dbc, BL, DTRANK + 2 * DSTATE, DIN, 0);
    // dt_proj: [8032,32] @ [1024,32]^T + bias, softplus   (TDM path, 1 chunk)
    k_gemm_tdm<<<dim3(DIN / 128, BL / 32), 256, 0, stream>>>(
        dbc, DTRANK + 2 * DSTATE, dt_proj_w + (size_t)layer * DIN * DTRANK,
        dt_proj_b + (size_t)layer * DIN, delta, DIN, DTRANK, 2);
    k_scan<<<dim3(DIN / 256, B_SZ), 256, 0, stream>>>(
        delta, u, dbc, A_log + (size_t)layer * DIN * DSTATE,
        Dp + (size_t)layer * DIN, ybuf);
    {
      int n = BL * DIN;
      k_gate<<<(n + 255) / 256, 256, 0, stream>>>(ybuf, xz);
    }
    // out_proj: [8032,1024] @ [512,1024]^T   (TDM path)
    k_gemm_tdm<<<dim3(DMODEL / 128, BL / 32), 256, 0, stream>>>(
        ybuf, DIN, out_proj_w + (size_t)layer * DMODEL * DIN, nullptr, outb,
        DMODEL, DIN, 0);
    k_resln<<<BL, 256, 0, stream>>>(x, outb, ln_g + (size_t)layer * DMODEL,
                                    ln_b + (size_t)layer * DMODEL);
  }
  {
    int n = B_SZ * DMODEL;
    k_pool<<<(n + 255) / 256, 256, 0, stream>>>(x, fbuf);
  }
  // fusion-1: M=32, N=256, K=544 (TDM path)
  k_gemm_tdm<<<dim3(256 / 128, B_SZ / 32), 256, 0, stream>>>(
      fbuf, DMODEL + 32, f_w1, f_b1, h1, 256, DMODEL + 32, 1);
  // fusion-2: N=64 -> direct WMMA path
  k_gemm_wmma<<<gemm_grid(B_SZ, 64), 256, 0, stream>>>(h1, 256, f_w2, f_b2, h2,
                                                       B_SZ, 64, 256, 1);
  k_final<<<1, 32, 0, stream>>>(h2, pdb, out_w, out_b, aux_w, aux_b, outp);
}